// MultiBoxLoss_59055800320339
// MI455X (gfx1250) — compile-verified
//
#include <hip/hip_runtime.h>
#include <hip/hip_bf16.h>
#include <stdint.h>

#define THRESH   0.5f
#define VAR0     0.1f
#define VAR1     0.2f
#define ALPHA    0.25f
#define NCLS     2
#define EPSF     1e-7f
#define MAXO     32
#define BLK      256

typedef float v2f __attribute__((ext_vector_type(2)));
typedef float v8f __attribute__((ext_vector_type(8)));

__device__ __forceinline__ float iou_corner(float ax0, float ay0, float ax1, float ay1,
                                            float bx0, float by0, float bx1, float by1) {
    float lx = fmaxf(ax0, bx0), ly = fmaxf(ay0, by0);
    float rx = fminf(ax1, bx1), ry = fminf(ay1, by1);
    float w  = fmaxf(rx - lx, 0.0f), h = fmaxf(ry - ly, 0.0f);
    float inter = w * h;
    float aa = (ax1 - ax0) * (ay1 - ay0);
    float ab = (bx1 - bx0) * (by1 - by0);
    return inter / (aa + ab - inter);
}

// Half-wave sums of x+y (lanes 0-15 -> col0, lanes 16-31 -> col1) using V_WMMA_F32_16X16X4_F32.
// A[m][k]: lanes 0-15 hold (K0,K1)=(x,y); lanes 16-31 hold (K2,K3)=(x,y).
// B selector: col0 = (1,1,0,0), col1 = (0,0,1,1) over K  =>
//   D[m][0] = x_m + y_m,  D[m][1] = x_{m+16} + y_{m+16}.
// Per-lane sum of the 8 C VGPRs + shfl_xor(16) completes the reduction:
// lanes {0,16} end with sum(x+y) over lanes 0-15, lanes {1,17} over lanes 16-31.
__device__ __forceinline__ float wave_sum2_wmma(float x, float y, unsigned lane) {
    v2f a; a.x = x; a.y = y;
    float bv = (lane == 0u || lane == 17u) ? 1.0f : 0.0f;
    v2f bb; bb.x = bv; bb.y = bv;
    v8f c = {0.f, 0.f, 0.f, 0.f, 0.f, 0.f, 0.f, 0.f};
    c = __builtin_amdgcn_wmma_f32_16x16x4_f32(false, a, false, bb, (short)0, c, false, false);
    float s = c[0] + c[1] + c[2] + c[3] + c[4] + c[5] + c[6] + c[7];
    s += __shfl_xor(s, 16, 32);
    return s;
}

// ---------------------------------------------------------------------------
// Kernel 0: zero the packed best-prior keys (must re-init every call).
__global__ void k_init(unsigned long long* __restrict__ packed, int n) {
    int i = blockIdx.x * blockDim.x + threadIdx.x;
    if (i < n) packed[i] = 0ull;
}

// ---------------------------------------------------------------------------
// Kernel A: per-(b,o) argmax over priors via packed u64 max.
// key = (bits(iou) << 32) | ~prior_idx   (iou >= 0 => monotone in float order;
// ~idx gives "first index wins" on ties, matching jnp.argmax).
__global__ void k_bestprior(const float* __restrict__ priors,
                            const float* __restrict__ tgt,     // [B,O,5]
                            unsigned long long* __restrict__ packed, // [B,O]
                            int P, int O) {
    __shared__ float sT[MAXO * 5];
    __shared__ unsigned long long sBest[MAXO];
    const int b   = blockIdx.y;
    const int tid = threadIdx.x;
    const int nT  = O * 5;
    for (int i = tid; i < nT; i += blockDim.x) sT[i] = tgt[(size_t)b * nT + i];
    for (int i = tid; i < O;  i += blockDim.x) sBest[i] = 0ull;
    __syncthreads();

    const int p  = blockIdx.x * blockDim.x + tid;
    const bool act = (p < P);
    const int pc = act ? p : 0;
    float4 pr = ((const float4*)priors)[pc];              // [cx,cy,w,h]
    float bx0 = pr.x - pr.z * 0.5f, by0 = pr.y - pr.w * 0.5f;
    float bx1 = pr.x + pr.z * 0.5f, by1 = pr.y + pr.w * 0.5f;

    if (act) {
        for (int o = 0; o < O; ++o) {
            const float* t = &sT[o * 5];
            float ov = iou_corner(t[0], t[1], t[2], t[3], bx0, by0, bx1, by1);
            unsigned long long key =
                ((unsigned long long)__float_as_uint(ov) << 32) |
                (unsigned long long)(~(unsigned)p);
            atomicMax(&sBest[o], key);                    // ds_max_u64
        }
    }
    __syncthreads();
    for (int o = tid; o < O; o += blockDim.x)
        atomicMax(&packed[(size_t)b * O + o], sBest[o]);  // global_atomic_max_u64
}

// ---------------------------------------------------------------------------
// Kernel B: per-prior match + smooth-L1 + focal; per-block partial sums.
__global__ void k_loss(const float* __restrict__ loc_data,   // [B,P,4]
                       const float* __restrict__ conf_data,  // [B,P,2]
                       const float* __restrict__ priors,     // [P,4]
                       const float* __restrict__ tgt,        // [B,O,5]
                       const unsigned long long* __restrict__ packed, // [B,O]
                       float* __restrict__ partials,         // [gridBlocks*3]
                       int P, int O) {
    __shared__ float sT[MAXO * 5];
    __shared__ int   sBP[MAXO];
    __shared__ float sWave[(BLK / 32) * 4];
    const int b   = blockIdx.y;
    const int tid = threadIdx.x;
    const int nT  = O * 5;

    // prefetch this block's loc/conf tiles (global_prefetch_b8)
    {
        size_t base = ((size_t)b * P + (size_t)blockIdx.x * blockDim.x);
        __builtin_prefetch(&loc_data[base * 4 + (size_t)tid * 4], 0, 1);
        __builtin_prefetch(&conf_data[base * 2 + (size_t)tid * 2], 0, 1);
    }

    for (int i = tid; i < nT; i += blockDim.x) sT[i] = tgt[(size_t)b * nT + i];
    for (int i = tid; i < O;  i += blockDim.x) {
        unsigned long long k = packed[(size_t)b * O + i];
        sBP[i] = (int)(~(unsigned)(k & 0xFFFFFFFFull));
    }
    __syncthreads();

    const int p   = blockIdx.x * blockDim.x + tid;
    const bool act = (p < P);
    const int pc  = act ? p : 0;

    float4 pr = ((const float4*)priors)[pc];
    float bx0 = pr.x - pr.z * 0.5f, by0 = pr.y - pr.w * 0.5f;
    float bx1 = pr.x + pr.z * 0.5f, by1 = pr.y + pr.w * 0.5f;

    // best truth per prior (strict '>' => first-max, matching jnp.argmax)
    float best = -1.0f; int bidx = 0;
    for (int o = 0; o < O; ++o) {
        const float* t = &sT[o * 5];
        float ov = iou_corner(t[0], t[1], t[2], t[3], bx0, by0, bx1, by1);
        if (ov > best) { best = ov; bidx = o; }
    }
    // forced-match override (forward loop => last write wins, like .at[].set)
    for (int o = 0; o < O; ++o)
        if (sBP[o] == p) { best = 2.0f; bidx = o; }

    const float* m = &sT[bidx * 5];
    int conf = (best < THRESH) ? 0 : ((int)m[4] + 1);

    float ll = 0.0f, lc = 0.0f, np = 0.0f;
    if (act && conf > 0) {
        float gx = ((m[0] + m[2]) * 0.5f - pr.x) / (VAR0 * pr.z);
        float gy = ((m[1] + m[3]) * 0.5f - pr.y) / (VAR0 * pr.w);
        float gw = logf((m[2] - m[0]) / pr.z) * (1.0f / VAR1);
        float gh = logf((m[3] - m[1]) / pr.w) * (1.0f / VAR1);
        float4 ld = ((const float4*)loc_data)[(size_t)b * P + p];
        float d0 = ld.x - gx, d1 = ld.y - gy, d2 = ld.z - gw, d3 = ld.w - gh;
        float a0 = fabsf(d0), a1 = fabsf(d1), a2 = fabsf(d2), a3 = fabsf(d3);
        ll += (a0 < 1.0f) ? 0.5f * d0 * d0 : a0 - 0.5f;
        ll += (a1 < 1.0f) ? 0.5f * d1 * d1 : a1 - 0.5f;
        ll += (a2 < 1.0f) ? 0.5f * d2 * d2 : a2 - 0.5f;
        ll += (a3 < 1.0f) ? 0.5f * d3 * d3 : a3 - 0.5f;
    }
    if (act) {
        np = (conf > 0) ? 1.0f : 0.0f;
        const float2 cd = ((const float2*)conf_data)[(size_t)b * P + pc];
        float mm = fmaxf(cd.x, cd.y);
        float e0 = expf(cd.x - mm), e1 = expf(cd.y - mm);
        float inv = 1.0f / (e0 + e1);
        float p0 = fminf(fmaxf(e0 * inv, EPSF), 1.0f - EPSF);
        float p1 = fminf(fmaxf(e1 * inv, EPSF), 1.0f - EPSF);
        if (conf < NCLS) {   // one-hot row is all-zero if conf >= NCLS
            float pt = (conf == 0) ? p0 : p1;
            float om = 1.0f - pt;
            lc = ALPHA * (-logf(pt)) * om * om;
        }
    }

    // --- wave reduction via V_WMMA_F32_16X16X4_F32 (convergent, full EXEC) ---
    const unsigned lane = tid & 31u;
    const unsigned wid  = tid >> 5;
    float s12 = wave_sum2_wmma(ll, lc, lane);   // lane0: sum(ll+lc) lanes 0-15, lane1: lanes 16-31
    float s3  = wave_sum2_wmma(np, 0.0f, lane); // lane0: sum(np) over lanes 0-15
    if (lane == 0u) { sWave[wid * 4 + 0] = s12; sWave[wid * 4 + 2] = s3; }
    if (lane == 1u) { sWave[wid * 4 + 1] = s12; }
    __syncthreads();
    if (tid == 0) {
        float tl = 0.f, tc = 0.f, tn = 0.f;
        const int nw = blockDim.x >> 5;
        for (int w = 0; w < nw; ++w) {
            tl += sWave[w * 4 + 0];
            tc += sWave[w * 4 + 1];
            tn += sWave[w * 4 + 2];
        }
        int bi = blockIdx.y * gridDim.x + blockIdx.x;
        partials[(size_t)bi * 3 + 0] = tl;
        partials[(size_t)bi * 3 + 1] = tc;
        partials[(size_t)bi * 3 + 2] = tn;
    }
}

// ---------------------------------------------------------------------------
// Kernel C: deterministic fixed-order final reduction.
__global__ void k_final(const float* __restrict__ partials, int nb,
                        float* __restrict__ out) {
    __shared__ float s[BLK * 3];
    const int tid = threadIdx.x;
    float tl = 0.f, tc = 0.f, tn = 0.f;
    for (int i = tid; i < nb; i += BLK) {
        tl += partials[(size_t)i * 3 + 0];
        tc += partials[(size_t)i * 3 + 1];
        tn += partials[(size_t)i * 3 + 2];
    }
    s[tid * 3 + 0] = tl; s[tid * 3 + 1] = tc; s[tid * 3 + 2] = tn;
    __syncthreads();
    for (int st = BLK / 2; st > 0; st >>= 1) {
        if (tid < st) {
            s[tid * 3 + 0] += s[(tid + st) * 3 + 0];
            s[tid * 3 + 1] += s[(tid + st) * 3 + 1];
            s[tid * 3 + 2] += s[(tid + st) * 3 + 2];
        }
        __syncthreads();
    }
    if (tid == 0) {
        float npos = fmaxf(1.0f, s[2]);
        out[0] = s[0] / npos;
        out[1] = s[1] / npos;
    }
}

// ---------------------------------------------------------------------------
extern "C" void kernel_launch(void* const* d_in, const int* in_sizes, int n_in,
                              void* d_out, int out_size, void* d_ws, size_t ws_size,
                              hipStream_t stream) {
    const float* loc    = (const float*)d_in[0]; // [B,P,4]
    const float* conf   = (const float*)d_in[1]; // [B,P,2]
    const float* priors = (const float*)d_in[2]; // [P,4]
    const float* tgt    = (const float*)d_in[3]; // [B,O,5]

    const int P = in_sizes[2] / 4;
    const int B = in_sizes[0] / (4 * P);
    int O = in_sizes[3] / (5 * B);
    if (O > MAXO) O = MAXO;

    const int PB = (P + BLK - 1) / BLK;
    const int nb = PB * B;

    unsigned long long* packed = (unsigned long long*)d_ws;
    size_t packed_bytes = ((size_t)B * O * sizeof(unsigned long long) + 255) & ~(size_t)255;
    float* partials = (float*)((char*)d_ws + packed_bytes);

    k_init<<<(B * O + BLK - 1) / BLK, BLK, 0, stream>>>(packed, B * O);

    dim3 grid(PB, B);
    k_bestprior<<<grid, BLK, 0, stream>>>(priors, tgt, packed, P, O);
    k_loss<<<grid, BLK, 0, stream>>>(loc, conf, priors, tgt, packed, partials, P, O);
    k_final<<<1, BLK, 0, stream>>>(partials, nb, (float*)d_out);
}